// GCN_13280038879718
// MI455X (gfx1250) — compile-verified
//
#include <hip/hip_runtime.h>
#include <hip/hip_bf16.h>

// GCN 3-layer forward for MI455X (gfx1250, wave32).
//   h  = X @ W.T                         (WMMA f32 16x16x4)
//   agg = segment_sum(h[src], dst)       (global_atomic_add_f32)
//   next layer input = relu(agg + b)     (fused into next GEMM's A-load,
//                                         selected at compile time)
// Workspace: bufA = N*128 f32, bufB = N*128 f32.

typedef float v2f __attribute__((ext_vector_type(2)));
typedef float v8f __attribute__((ext_vector_type(8)));

#define K_DIM   128          // input feature dim for every layer
#define LDSP    132          // padded LDS row stride (banks: 4*j + koff, conflict-free)

// ---------------------------------------------------------------------------
// GEMM: H[m, j0_base + j] = sum_k act(Xin[m,k]) * W[j0_base + j, k]
//   act(v) = BIAS ? relu(v + bias[k]) : v        (compile-time selection)
// Block: 256 threads = 8 waves; wave w -> M-tile rows [blockIdx.x*128 + 16w, +16)
// blockIdx.y -> j0_base = blockIdx.y * NJ * 16   (NJ 16-wide output tiles)
// ---------------------------------------------------------------------------
template <int NJ, bool BIAS>
__global__ __launch_bounds__(256)
void gcn_gemm_xwT(const float* __restrict__ Xin,
                  const float* __restrict__ W,
                  const float* __restrict__ bias,   // input-side bias (+relu); used iff BIAS
                  float* __restrict__ H,
                  int Nn, int Fout)
{
    __shared__ float sW[NJ * 16 * LDSP];

    const int tid = threadIdx.x;
    const int j0_base = blockIdx.y * (NJ * 16);

    // Cooperative load of W rows [j0_base, j0_base + NJ*16) into padded LDS.
    // Rows beyond Fout are zero-filled (their D columns are never stored).
    for (int i = tid; i < NJ * 16 * (K_DIM / 4); i += 256) {
        const int row = i >> 5;              // K_DIM/4 == 32
        const int c4  = (i & 31) << 2;
        float4 v = make_float4(0.f, 0.f, 0.f, 0.f);
        const int grow = j0_base + row;
        if (grow < Fout) v = *(const float4*)(W + (long)grow * K_DIM + c4);
        float* dst = &sW[row * LDSP + c4];
        dst[0] = v.x; dst[1] = v.y; dst[2] = v.z; dst[3] = v.w;
    }
    __syncthreads();

    const int wave  = tid >> 5;
    const int lane  = tid & 31;
    const int jl    = lane & 15;          // A row-in-tile / B col-in-tile
    const int khalf = (lane >> 4) << 1;   // 0 or 2 (K sub-offset per ISA layout)

    const int m0 = blockIdx.x * 128 + wave * 16;
    int m = m0 + jl;
    if (m >= Nn) m = Nn - 1;              // clamp loads; stores are guarded
    const float* __restrict__ xrow = Xin + (long)m * K_DIM;

    v8f acc[NJ];
#pragma unroll
    for (int j = 0; j < NJ; ++j) { v8f z = {}; acc[j] = z; }

#pragma unroll
    for (int kb = 0; kb < K_DIM; kb += 4) {
        const int kk = kb + khalf;
        v2f a;
        a.x = xrow[kk];
        a.y = xrow[kk + 1];
        if (BIAS) {   // compile-time: no runtime branch in either instantiation
            a.x = fmaxf(a.x + bias[kk],     0.0f);
            a.y = fmaxf(a.y + bias[kk + 1], 0.0f);
        }
#pragma unroll
        for (int j = 0; j < NJ; ++j) {
            const float* wp = &sW[(j * 16 + jl) * LDSP + kk];
            v2f b;
            b.x = wp[0];
            b.y = wp[1];
            // D = A(16x4) * B(4x16) + C   (fp32 WMMA, wave32)
            acc[j] = __builtin_amdgcn_wmma_f32_16x16x4_f32(
                false, a, false, b, (short)0, acc[j], false, false);
        }
    }

    // D layout: lane L, VGPR r -> row M = r + 8*(L/16), col N = L%16
    const int rbase = m0 + ((lane >> 4) << 3);
#pragma unroll
    for (int j = 0; j < NJ; ++j) {
        const int col = j0_base + j * 16 + jl;
        if (col < Fout) {
#pragma unroll
            for (int r = 0; r < 8; ++r) {
                const int row = rbase + r;
                if (row < Nn) H[(long)row * Fout + col] = acc[j][r];
            }
        }
    }
}

// ---------------------------------------------------------------------------
// Scatter-add: out[dst[e], :] += h[src[e], :]   (float4 gather, f32 atomics)
// ---------------------------------------------------------------------------
template <int F4, int F>
__global__ __launch_bounds__(256)
void gcn_scatter_add(const float* __restrict__ h,
                     const int* __restrict__ src,
                     const int* __restrict__ dst,
                     float* __restrict__ out,
                     int E)
{
    const long idx   = (long)blockIdx.x * blockDim.x + threadIdx.x;
    const long total = (long)E * F4;
    if (idx >= total) return;
    const int e = (int)(idx / F4);
    const int f = (int)(idx % F4) << 2;
    const int s = src[e];
    const int d = dst[e];
    const float4 v = *(const float4*)(h + (long)s * F + f);
    float* o = out + (long)d * F + f;
    atomicAdd(o + 0, v.x);
    atomicAdd(o + 1, v.y);
    atomicAdd(o + 2, v.z);
    atomicAdd(o + 3, v.w);
}

// out[i*F + j] = b[j]  (seed the final output with b2 before the last scatter)
template <int F>
__global__ __launch_bounds__(256)
void gcn_fill_bias(float* __restrict__ out, const float* __restrict__ b, long n)
{
    const long i = (long)blockIdx.x * blockDim.x + threadIdx.x;
    if (i < n) out[i] = b[(int)(i % F)];
}

extern "C" void kernel_launch(void* const* d_in, const int* in_sizes, int n_in,
                              void* d_out, int out_size, void* d_ws, size_t ws_size,
                              hipStream_t stream)
{
    const float* x  = (const float*)d_in[0];
    const int*   ei = (const int*)d_in[1];      // (2, E): [src | dst]
    const float* W0 = (const float*)d_in[2];
    const float* b0 = (const float*)d_in[3];
    const float* W1 = (const float*)d_in[4];
    const float* b1 = (const float*)d_in[5];
    const float* W2 = (const float*)d_in[6];
    const float* b2 = (const float*)d_in[7];

    const int Nn   = in_sizes[0] / K_DIM;       // 50000
    const int E    = in_sizes[1] / 2;           // 800000
    const int Fout = in_sizes[7];               // 40
    const int* srcI = ei;
    const int* dstI = ei + E;

    float* bufA = (float*)d_ws;                 // N x 128
    float* bufB = bufA + (size_t)Nn * K_DIM;    // N x 128
    float* outf = (float*)d_out;                // N x 40

    const int mblocks = (Nn + 127) / 128;
    const dim3 blk(256);

    // Layer 0: h0 = x @ W0.T
    gcn_gemm_xwT<4, false><<<dim3(mblocks, 2), blk, 0, stream>>>(x, W0, nullptr, bufA, Nn, K_DIM);
    // agg0 = scatter(h0)
    hipMemsetAsync(bufB, 0, (size_t)Nn * K_DIM * sizeof(float), stream);
    {
        const long total = (long)E * 32;
        gcn_scatter_add<32, 128><<<dim3((unsigned)((total + 255) / 256)), blk, 0, stream>>>(
            bufA, srcI, dstI, bufB, E);
    }
    // Layer 1: h1 = relu(agg0 + b0) @ W1.T
    gcn_gemm_xwT<4, true><<<dim3(mblocks, 2), blk, 0, stream>>>(bufB, W1, b0, bufA, Nn, K_DIM);
    // agg1 = scatter(h1)
    hipMemsetAsync(bufB, 0, (size_t)Nn * K_DIM * sizeof(float), stream);
    {
        const long total = (long)E * 32;
        gcn_scatter_add<32, 128><<<dim3((unsigned)((total + 255) / 256)), blk, 0, stream>>>(
            bufA, srcI, dstI, bufB, E);
    }
    // Layer 2: h2 = relu(agg1 + b1) @ W2.T   (Fout = 40, 3 j-tiles padded to 48)
    gcn_gemm_xwT<3, true><<<dim3(mblocks, 1), blk, 0, stream>>>(bufB, W2, b1, bufA, Nn, Fout);
    // out = b2 + scatter(h2)
    {
        const long n = (long)Nn * Fout;
        gcn_fill_bias<40><<<dim3((unsigned)((n + 255) / 256)), blk, 0, stream>>>(outf, b2, n);
        const long total = (long)E * 10;
        gcn_scatter_add<10, 40><<<dim3((unsigned)((total + 255) / 256)), blk, 0, stream>>>(
            bufA, srcI, dstI, outf, E);
    }
}